// CustomMoERouter_18803366822022
// MI455X (gfx1250) — compile-verified
//
#include <hip/hip_runtime.h>
#include <math.h>

// ---------------------------------------------------------------------------
// MoE sigmoid router for MI455X (gfx1250):
//   logits = X[16384,2048] @ W[2048,64] + b  (fp32 WMMA 16x16x4)
//   probs  = sigmoid(logits); top-8; normalize; dense routing map.
// ---------------------------------------------------------------------------

#define HIDDEN      2048
#define NEXPERTS    64
#define TOPK        8
#define TOKENS      16384          // 4 * 4096
#define BLK_TOKENS  256            // tokens per block (8 waves * 32 tokens)
#define THREADS     256
#define CHUNK_K     512            // K-chunk staged in LDS
#define WSTRIDE     516            // LDS row stride (dwords); 516 % 64 == 4 -> conflict-free b64 reads
#define PSTRIDE     68             // probs LDS row stride (dwords); multiple of 4 for float4 stores
#define SMEM_BYTES  (NEXPERTS * WSTRIDE * 4)   // 132096 B (>= 256*PSTRIDE*4 = 69632 B)

typedef float v2f __attribute__((ext_vector_type(2)));
typedef float v8f __attribute__((ext_vector_type(8)));

__global__ void moe_router_kernel(const float* __restrict__ X,
                                  const float* __restrict__ W,
                                  const float* __restrict__ bias,
                                  float* __restrict__ out_probs,   // [TOKENS][8]
                                  float* __restrict__ out_idx,     // [TOKENS][8] (as float)
                                  float* __restrict__ out_map) {   // [TOKENS][64]
    extern __shared__ float smem[];   // phase 1: W chunk (transposed); phase 2: probs

    const int tid  = threadIdx.x;
    const int wave = tid >> 5;
    const int lane = tid & 31;
    const int l15  = lane & 15;
    const int kh   = lane >> 4;          // K-half selector for A/B operands

    const int blk_tok0  = blockIdx.x * BLK_TOKENS;
    const int wave_tok0 = blk_tok0 + wave * 32;

    // A operand rows for the two 16-token M-tiles this wave owns
    const float* xrow0 = X + (size_t)(wave_tok0 + l15)      * HIDDEN + 2 * kh;
    const float* xrow1 = X + (size_t)(wave_tok0 + 16 + l15) * HIDDEN + 2 * kh;

    // Per-lane base into the transposed W chunk (expert row), offset by K-half
    const float* bbase = smem + l15 * WSTRIDE + 2 * kh;

    v8f acc[2][4];
    #pragma unroll
    for (int t = 0; t < 2; ++t)
        #pragma unroll
        for (int n = 0; n < 4; ++n)
            acc[t][n] = (v8f){0.f, 0.f, 0.f, 0.f, 0.f, 0.f, 0.f, 0.f};

    // ---------------- GEMM over K in LDS-staged chunks ----------------
    for (int kc = 0; kc < HIDDEN; kc += CHUNK_K) {
        __syncthreads();   // previous chunk fully consumed
        // Cooperative transposed stage: smem[e*WSTRIDE + kr] = W[(kc+kr)*64 + e]
        for (int i4 = tid; i4 < (CHUNK_K * NEXPERTS) / 4; i4 += THREADS) {
            const int flat = i4 * 4;
            const int e    = flat & (NEXPERTS - 1);   // multiple of 4
            const int kr   = flat >> 6;
            const float4 w4 = *(const float4*)(W + (size_t)(kc + kr) * NEXPERTS + e);
            smem[(e + 0) * WSTRIDE + kr] = w4.x;
            smem[(e + 1) * WSTRIDE + kr] = w4.y;
            smem[(e + 2) * WSTRIDE + kr] = w4.z;
            smem[(e + 3) * WSTRIDE + kr] = w4.w;
        }
        __syncthreads();

        for (int kk = 0; kk < CHUNK_K; kk += 4) {
            const int k = kc + kk;
            const v2f a0 = *(const v2f*)(xrow0 + k);
            const v2f a1 = *(const v2f*)(xrow1 + k);
            // Load all 4 B operands into distinct registers first so the
            // scheduler can batch the ds loads and hide their latency
            // under the 8 WMMAs (instead of a dscnt-0 wait per pair).
            v2f bm[4];
            #pragma unroll
            for (int n = 0; n < 4; ++n)
                bm[n] = *(const v2f*)(bbase + n * 16 * WSTRIDE + kk);
            #pragma unroll
            for (int n = 0; n < 4; ++n) {
                acc[0][n] = __builtin_amdgcn_wmma_f32_16x16x4_f32(
                    false, a0, false, bm[n], (short)0, acc[0][n], false, false);
                acc[1][n] = __builtin_amdgcn_wmma_f32_16x16x4_f32(
                    false, a1, false, bm[n], (short)0, acc[1][n], false, false);
            }
        }
    }

    // ---------------- bias + sigmoid -> probs LDS [256][PSTRIDE] ----------------
    __syncthreads();   // done with W chunk buffer; reuse smem for probs
    float bv[4];
    #pragma unroll
    for (int n = 0; n < 4; ++n) bv[n] = bias[n * 16 + l15];

    #pragma unroll
    for (int t = 0; t < 2; ++t) {
        #pragma unroll
        for (int n = 0; n < 4; ++n) {
            const v8f c = acc[t][n];
            #pragma unroll
            for (int r = 0; r < 8; ++r) {
                const int ltok = wave * 32 + t * 16 + r + 8 * kh;  // token within block
                const int e    = n * 16 + l15;
                const float logit = c[r] + bv[n];
                smem[ltok * PSTRIDE + e] = 1.0f / (1.0f + __expf(-logit));
            }
        }
    }
    __syncthreads();

    // ---------------- per-token top-8, normalize, routing map ----------------
    {
        float* row = smem + tid * PSTRIDE;
        const int gtok = blk_tok0 + tid;
        float topv[TOPK];
        int   topi[TOPK];
        float sum = 0.0f;
        #pragma unroll
        for (int j = 0; j < TOPK; ++j) {
            float best = -1.0f;
            int   bi   = 0;
            for (int e = 0; e < NEXPERTS; ++e) {
                const float v = row[e];
                if (v > best) { best = v; bi = e; }
            }
            row[bi] = -best;          // tag as chosen, keep magnitude (sigmoid > 0)
            topv[j] = best;
            topi[j] = bi;
            sum += best;
        }
        const float rinv = 1.0f / sum;
        #pragma unroll
        for (int j = 0; j < TOPK; ++j) {
            out_probs[(size_t)gtok * TOPK + j] = topv[j] * rinv;
            out_idx  [(size_t)gtok * TOPK + j] = (float)topi[j];
        }
        // rewrite row as dense routing map
        for (int e = 0; e < NEXPERTS; ++e) {
            const float v = row[e];
            row[e] = (v < 0.0f) ? (-v) * rinv : 0.0f;
        }
    }
    __syncthreads();

    // Coalesced float4 store of routing map: 256*64 floats per block
    {
        float* dst = out_map + (size_t)blk_tok0 * NEXPERTS;
        for (int i4 = tid; i4 < (BLK_TOKENS * NEXPERTS) / 4; i4 += THREADS) {
            const int tok = i4 >> 4;            // (i4*4)/64
            const int e   = (i4 & 15) * 4;      // (i4*4)%64
            *(float4*)(dst + i4 * 4) = *(const float4*)(smem + tok * PSTRIDE + e);
        }
    }
}

extern "C" void kernel_launch(void* const* d_in, const int* in_sizes, int n_in,
                              void* d_out, int out_size, void* d_ws, size_t ws_size,
                              hipStream_t stream) {
    (void)in_sizes; (void)n_in; (void)d_ws; (void)ws_size; (void)out_size;
    const float* X = (const float*)d_in[0];   // [4,4096,2048]
    const float* W = (const float*)d_in[1];   // [2048,64]
    const float* b = (const float*)d_in[2];   // [64]

    float* out_probs = (float*)d_out;                               // 16384*8
    float* out_idx   = (float*)d_out + (size_t)TOKENS * TOPK;       // 16384*8
    float* out_map   = (float*)d_out + (size_t)2 * TOKENS * TOPK;   // 16384*64

    dim3 grid(TOKENS / BLK_TOKENS);   // 64 blocks
    dim3 block(THREADS);              // 8 waves (wave32)
    hipLaunchKernelGGL(moe_router_kernel, grid, block, SMEM_BYTES, stream,
                       X, W, b, out_probs, out_idx, out_map);
}